// GNNEncoder_49675591745971
// MI455X (gfx1250) — compile-verified
//
#include <hip/hip_runtime.h>

typedef __attribute__((ext_vector_type(2))) float v2f;
typedef __attribute__((ext_vector_type(8))) float v8f;

#define G_GROUPS 128
#define NPTS     1024
#define FDIM     16
#define KNN      16
#define HDIM     64
#define ROWS_PER_BLOCK 512
#define TILE_ROWS 32
#define DIST_STRIDE 1025

// Dynamic LDS layout (bytes)
#define OFF_XS     0        // 1024*16 f32      = 65536
#define OFF_SQ     65536    // 1024 f32         = 4096   (reused as Sacc in phase D)
#define OFF_IDX    69632    // 512*16 u16       = 16384
#define OFF_REGION 86016    // max(32*1025*4=131200, 1024*64*2=131072) = 131200
#define OFF_WD     217216   // 16*64 f32        = 4096
#define OFF_B1     221312   // 64 f32           = 256
#define SMEM_BYTES 221568

static __device__ __forceinline__ unsigned ordered_u32(float f) {
  unsigned u = __float_as_uint(f);
  unsigned m = ((unsigned)(((int)u) >> 31)) | 0x80000000u; // neg -> all ones, pos -> sign bit
  return u ^ m;
}

static __device__ __forceinline__ unsigned long long shfl_xor_u64(unsigned long long v, int m) {
  unsigned lo = (unsigned)v, hi = (unsigned)(v >> 32);
  lo = __shfl_xor(lo, m, 32);
  hi = __shfl_xor(hi, m, 32);
  return (((unsigned long long)hi) << 32) | lo;
}

__global__ __launch_bounds__(1024)
void gnn_group_kernel(const float* __restrict__ x, const float* __restrict__ W1,
                      const float* __restrict__ b1, float* __restrict__ partials) {
  extern __shared__ char smem[];
  float*          xs    = (float*)(smem + OFF_XS);
  float*          sq    = (float*)(smem + OFF_SQ);
  unsigned short* idxs  = (unsigned short*)(smem + OFF_IDX);
  unsigned*       distu = (unsigned*)(smem + OFF_REGION);   // phase B
  _Float16*       Bs    = (_Float16*)(smem + OFF_REGION);   // phase C/D (time-shared)
  float*          Wd    = (float*)(smem + OFF_WD);
  float*          b1s   = (float*)(smem + OFF_B1);
  float*          Sacc  = sq;                               // reuse sq region in phase D

  const int tid  = threadIdx.x;
  const int wave = tid >> 5;
  const int lane = tid & 31;
  const int lm   = lane & 15;
  const int hh   = lane >> 4;
  const int g    = blockIdx.x >> 1;
  const int half = blockIdx.x & 1;
  const int rowsBase = half * ROWS_PER_BLOCK;

  // ---- Phase A: load group's points into LDS, compute squared norms ----
  {
    const float* xp = x + ((size_t)g * NPTS + tid) * FDIM;
    float4 v0 = ((const float4*)xp)[0];
    float4 v1 = ((const float4*)xp)[1];
    float4 v2 = ((const float4*)xp)[2];
    float4 v3 = ((const float4*)xp)[3];
    float s = v0.x*v0.x + v0.y*v0.y + v0.z*v0.z + v0.w*v0.w
            + v1.x*v1.x + v1.y*v1.y + v1.z*v1.z + v1.w*v1.w
            + v2.x*v2.x + v2.y*v2.y + v2.z*v2.z + v2.w*v2.w
            + v3.x*v3.x + v3.y*v3.y + v3.z*v3.z + v3.w*v3.w;
    float4* xr = (float4*)(xs + tid * FDIM);
    xr[0] = v0; xr[1] = v1; xr[2] = v2; xr[3] = v3;
    sq[tid] = s;
  }
  __syncthreads();

  // ---- Phase B: per 32-row tile: WMMA Gram matrix -> ordered keys, then 16x argmin ----
  for (int tile = 0; tile < ROWS_PER_BLOCK / TILE_ROWS; ++tile) {
    const int tileBase = tile * TILE_ROWS;

    // B1: distances via chained V_WMMA_F32_16X16X4_F32 (K=16 as 4x K=4)
    {
      const int rowblk = wave >> 4;                       // 0..1 (16-row blk in tile)
      const float* ar = xs + (rowsBase + tileBase + rowblk * 16 + lm) * FDIM;
      v2f a0 = *(const v2f*)(ar + 0  + 2 * hh);
      v2f a1 = *(const v2f*)(ar + 4  + 2 * hh);
      v2f a2 = *(const v2f*)(ar + 8  + 2 * hh);
      v2f a3 = *(const v2f*)(ar + 12 + 2 * hh);
#pragma unroll
      for (int t = 0; t < 4; ++t) {
        const int colblk = (wave & 15) * 4 + t;           // 0..63
        const int j = colblk * 16 + lm;                   // this lane's column
        const float* br = xs + j * FDIM;
        v2f bb0 = *(const v2f*)(br + 0  + 2 * hh);
        v2f bb1 = *(const v2f*)(br + 4  + 2 * hh);
        v2f bb2 = *(const v2f*)(br + 8  + 2 * hh);
        v2f bb3 = *(const v2f*)(br + 12 + 2 * hh);
        v8f c = {0.f,0.f,0.f,0.f,0.f,0.f,0.f,0.f};
        c = __builtin_amdgcn_wmma_f32_16x16x4_f32(false, a0, false, bb0, (short)0, c, false, false);
        c = __builtin_amdgcn_wmma_f32_16x16x4_f32(false, a1, false, bb1, (short)0, c, false, false);
        c = __builtin_amdgcn_wmma_f32_16x16x4_f32(false, a2, false, bb2, (short)0, c, false, false);
        c = __builtin_amdgcn_wmma_f32_16x16x4_f32(false, a3, false, bb3, (short)0, c, false, false);
        const float sj = sq[j];
#pragma unroll
        for (int r = 0; r < 8; ++r) {
          const int m = rowblk * 16 + r + 8 * hh;         // row within 32-row tile
          const float key = sj - 2.0f * c[r];             // row-const sq_i dropped (rank-invariant)
          distu[m * DIST_STRIDE + j] = ordered_u32(key);
        }
      }
    }
    __syncthreads();

    // B2: one row per wave; 16x (scan + butterfly argmin), tie -> lowest j (matches top_k)
    {
      unsigned* drow = distu + wave * DIST_STRIDE;
      const int outRow = tileBase + wave;                 // 0..511
      for (int it = 0; it < KNN; ++it) {
        unsigned long long best = ~0ull;
#pragma unroll 8
        for (int s = 0; s < 32; ++s) {
          const int j = lane + 32 * s;
          const unsigned long long k =
              (((unsigned long long)drow[j]) << 32) | (unsigned)j;
          best = (k < best) ? k : best;
        }
#pragma unroll
        for (int off = 16; off > 0; off >>= 1) {
          const unsigned long long o = shfl_xor_u64(best, off);
          best = (o < best) ? o : best;
        }
        const unsigned bj = (unsigned)best;
        if (lane == 0) {
          idxs[outRow * KNN + it] = (unsigned short)bj;
          drow[bj] = 0xFFFFFFFFu;                         // exclude (DS ops in-order per wave)
        }
      }
    }
    __syncthreads();
  }

  // ---- Phase C: Wd = W1_top - W1_bot (+b1) to LDS; B-table = X @ W1_bot via WMMA, f16 in LDS ----
  Wd[tid] = W1[tid] - W1[tid + FDIM * HDIM];              // 16x64 elements, one per thread
  if (tid < HDIM) b1s[tid] = b1[tid];
  __syncthreads();
  for (int t = 0; t < 8; ++t) {
    const int tileId = wave * 8 + t;                      // 256 tiles: 64 rowblk x 4 colblk
    const int rowblk = tileId >> 2;
    const int colblk = tileId & 3;
    const float* ar = xs + (rowblk * 16 + lm) * FDIM;
    const int col = colblk * 16 + lm;
    v8f c = {0.f,0.f,0.f,0.f,0.f,0.f,0.f,0.f};
#pragma unroll
    for (int kk = 0; kk < 4; ++kk) {
      v2f a = *(const v2f*)(ar + 4 * kk + 2 * hh);
      const int krow = FDIM + 4 * kk + 2 * hh;            // row of W1_bot
      v2f b;
      b.x = W1[krow * HDIM + col];
      b.y = W1[(krow + 1) * HDIM + col];
      c = __builtin_amdgcn_wmma_f32_16x16x4_f32(false, a, false, b, (short)0, c, false, false);
    }
#pragma unroll
    for (int r = 0; r < 8; ++r) {
      const int row = rowblk * 16 + r + 8 * hh;
      Bs[row * HDIM + col] = (_Float16)c[r];
    }
  }
  __syncthreads();

  // ---- Phase D: S = sum_{i,k} relu(A_i + B_{idx}) ; lane owns dims (lane, lane+32) ----
  if (tid < HDIM) Sacc[tid] = 0.f;
  __syncthreads();
  {
    float s0 = 0.f, s1 = 0.f;
    for (int it = 0; it < ROWS_PER_BLOCK / 32; ++it) {
      const int r = wave + 32 * it;                       // 0..511
      const float* xr = xs + (rowsBase + r) * FDIM;
      float a0 = b1s[lane], a1 = b1s[lane + 32];
#pragma unroll
      for (int f = 0; f < FDIM; ++f) {
        const float xf = xr[f];
        a0 += xf * Wd[f * HDIM + lane];
        a1 += xf * Wd[f * HDIM + lane + 32];
      }
#pragma unroll
      for (int k = 0; k < KNN; ++k) {
        const int j = idxs[r * KNN + k];
        const _Float16* bp = Bs + j * HDIM;
        const float h0 = a0 + (float)bp[lane];
        const float h1 = a1 + (float)bp[lane + 32];
        s0 += fmaxf(h0, 0.f);
        s1 += fmaxf(h1, 0.f);
      }
    }
    atomicAdd(&Sacc[lane], s0);
    atomicAdd(&Sacc[lane + 32], s1);
  }
  __syncthreads();
  if (tid < HDIM) partials[(size_t)blockIdx.x * HDIM + tid] = Sacc[tid];
}

// Head: out[g] = ((mean S)@W2 + b2)@Wf + bf   (tiny: 128x64x64 + 128x64x128)
__global__ __launch_bounds__(128)
void gnn_head_kernel(const float* __restrict__ P, const float* __restrict__ W2,
                     const float* __restrict__ b2, const float* __restrict__ Wf,
                     const float* __restrict__ bf, float* __restrict__ out) {
  __shared__ float Sm[HDIM];
  __shared__ float T[HDIM];
  const int g = blockIdx.x, tid = threadIdx.x;
  if (tid < HDIM) {
    Sm[tid] = (P[(size_t)(2 * g) * HDIM + tid] + P[(size_t)(2 * g + 1) * HDIM + tid])
              * (1.0f / (float)(NPTS * KNN));
  }
  __syncthreads();
  if (tid < HDIM) {
    float acc = b2[tid];
#pragma unroll
    for (int h = 0; h < HDIM; ++h) acc += Sm[h] * W2[h * HDIM + tid];
    T[tid] = acc;
  }
  __syncthreads();
  float acc = bf[tid];
#pragma unroll
  for (int l = 0; l < HDIM; ++l) acc += T[l] * Wf[l * 128 + tid];
  out[(size_t)g * 128 + tid] = acc;
}

extern "C" void kernel_launch(void* const* d_in, const int* in_sizes, int n_in,
                              void* d_out, int out_size, void* d_ws, size_t ws_size,
                              hipStream_t stream) {
  (void)in_sizes; (void)n_in; (void)out_size; (void)ws_size;
  const float* x  = (const float*)d_in[0];
  // d_in[1] = batch (int) -- unused: reference reshapes by fixed group size
  const float* W1 = (const float*)d_in[2];
  const float* b1 = (const float*)d_in[3];
  const float* W2 = (const float*)d_in[4];
  const float* b2 = (const float*)d_in[5];
  const float* Wf = (const float*)d_in[6];
  const float* bf = (const float*)d_in[7];
  float* out = (float*)d_out;
  float* partials = (float*)d_ws;                          // 256*64 f32 = 64 KB

  gnn_group_kernel<<<dim3(G_GROUPS * 2), dim3(1024), SMEM_BYTES, stream>>>(x, W1, b1, partials);
  gnn_head_kernel<<<dim3(G_GROUPS), dim3(128), 0, stream>>>(partials, W2, b2, Wf, bf, out);
}